// ModulatedDeformableConv2dResoff_82119774699852
// MI455X (gfx1250) — compile-verified
//
#include <hip/hip_runtime.h>
#include <hip/hip_bf16.h>
#include <math.h>

// Problem constants (from reference)
#define Bv     4
#define CINv   128
#define COUTv  128
#define Kv     3
#define K2v    9
#define Hv     96
#define Wv     96
#define HWv    (Hv * Wv)          // 9216
#define REDv   (CINv * K2v)       // 1152 reduction length
#define MASK_SCALE_F 2.0f

#define HALF_RED 576              // 64 channels * 9 taps

typedef __attribute__((ext_vector_type(2))) float v2f;
typedef __attribute__((ext_vector_type(8))) float v8f;

// ---------------------------------------------------------------------------
// Kernel 1: offset/mask conv as im2col + fp32 WMMA GEMM.
//   om[co,p] = sum_r w_offmask[co][r] * P[r][p],  co in [0,27) padded to 32.
// Block = 256 threads (8 waves) per 16-pixel tile. Wave w: co-tile (w&1),
// K-slice (w>>1) of 144 rows per half. Partials reduced through LDS.
// ---------------------------------------------------------------------------
__global__ void offmask_wmma_kernel(const float* __restrict__ x,
                                    const float* __restrict__ w_offmask, // [27][1152]
                                    float* __restrict__ offset_out,      // [B][18][H][W]
                                    float* __restrict__ mask_out)        // [B][9][H][W]
{
    __shared__ float sP[HALF_RED * 16];    // im2col panel half: 576 x 16 (36 KB)
    __shared__ float sRed[2 * 4 * 32 * 9]; // partials [tile][slice][lane][8], pad 9

    int lin = blockIdx.x;
    const int wt = lin % (Wv / 16); lin /= (Wv / 16);
    const int ho = lin % Hv;        lin /= Hv;
    const int b  = lin;
    const int woBase = wt * 16;

    const int tid  = threadIdx.x;
    const int lane = tid & 31;
    const int wave = tid >> 5;

    const float* __restrict__ xb = x + (size_t)b * CINv * HWv;

    // WMMA fragment mapping (ISA 7.12.2, fp32 16x16x4):
    //   A: lane L -> row m=L%16, cols 2*(L/16)+{0,1}
    //   B: lane L -> col n=L%16, rows 2*(L/16)+{0,1}
    const int m16    = lane & 15;
    const int khalf  = (lane >> 4) << 1;
    const int tileT  = wave & 1;          // co-tile: rows [16t, 16t+16)
    const int kslice = wave >> 1;         // 0..3 -> 144-row slice within each half

    const int coRow     = tileT * 16 + m16;
    const int coClamped = coRow < 27 ? coRow : 26;   // padded rows read row 26, discarded
    const float* __restrict__ wrow =
        w_offmask + (size_t)coClamped * REDv + khalf;

    v8f acc = {};

    for (int half = 0; half < 2; ++half) {
        if (half) __syncthreads();       // WAR on sP

        // ---- im2col fill: 64 channels x 9 taps x 16 pixels, zero-padded ----
        for (int e = tid; e < 64 * 144; e += 256) {
            const int cl = e / 144;
            const int q  = e % 144;
            const int c  = half * 64 + cl;
            const int k2 = q >> 4, p = q & 15;
            const int ki = k2 / 3, kj = k2 % 3;
            const int y  = ho - 1 + ki;
            const int xx = woBase + p - 1 + kj;
            const bool v = (y >= 0) && (y < Hv) && (xx >= 0) && (xx < Wv);
            sP[(cl * 9 + k2) * 16 + p] =
                v ? xb[(size_t)c * HWv + y * Wv + xx] : 0.0f;
        }
        __syncthreads();

        // ---- 36 fp32 WMMAs over this wave's 144-row K-slice ----
        const float* __restrict__ wrh  = wrow + half * HALF_RED + kslice * 144;
        const float* __restrict__ pbas = &sP[(kslice * 144) * 16];
        for (int kk = 0; kk < 144; kk += 4) {
            v2f a, bf;
            a.x  = wrh[kk];
            a.y  = wrh[kk + 1];
            bf.x = pbas[(kk + khalf)     * 16 + m16];
            bf.y = pbas[(kk + khalf + 1) * 16 + m16];
            acc = __builtin_amdgcn_wmma_f32_16x16x4_f32(
                false, a, false, bf, (short)0, acc, false, false);
        }
    }

    // ---- cross-wave K-slice reduction through LDS ----
    __syncthreads();                     // sP no longer needed
#pragma unroll
    for (int v = 0; v < 8; ++v)
        sRed[((tileT * 4 + kslice) * 32 + lane) * 9 + v] = acc[v];
    __syncthreads();

    if (wave < 2) {
        const int t  = wave;
        const int hi = lane >> 4;
        const int wo = woBase + m16;
#pragma unroll
        for (int v = 0; v < 8; ++v) {
            float s = 0.0f;
#pragma unroll
            for (int sl = 0; sl < 4; ++sl)
                s += sRed[((t * 4 + sl) * 32 + lane) * 9 + v];
            const int co = t * 16 + (v + 8 * hi);
            if (co < 18) {
                offset_out[(((size_t)b * 18 + co) * Hv + ho) * Wv + wo] = s;
            } else if (co < 27) {
                mask_out[(((size_t)b * 9 + (co - 18)) * Hv + ho) * Wv + wo] =
                    MASK_SCALE_F / (1.0f + __expf(-s));
            }
        }
    }
}

// ---------------------------------------------------------------------------
// Kernel 2: bilinear deformable sampling (B-matrix into LDS) + fp32 WMMA GEMM.
//   out[b,co,p] = sum_{r=c*9+k2} weight[co][r] * sampled[r][p] + bias[co]
// One block = 256 threads = 8 waves; one 16-pixel tile (along W) per block.
// Wave w computes co in [16w, 16w+16). Reduction split into two CIN halves.
// ---------------------------------------------------------------------------
__global__ void deform_wmma_kernel(const float* __restrict__ x,
                                   const float* __restrict__ weight,   // [COUT][1152]
                                   const float* __restrict__ bias,     // [COUT]
                                   const float* __restrict__ offsets,  // [B][18][H][W]
                                   const float* __restrict__ mask,     // [B][9][H][W]
                                   float* __restrict__ out)            // [B][COUT][H][W]
{
    __shared__ float sB[HALF_RED * 16];   // B-panel: 576 rows x 16 pixels (36 KB)
    __shared__ int   sIdx[4 * 144];       // 4 bilinear corners per (k2,pixel)
    __shared__ float sWgt[4 * 144];       // corner weight * validity * mask

    int lin = blockIdx.x;
    const int wt = lin % (Wv / 16); lin /= (Wv / 16);
    const int ho = lin % Hv;        lin /= Hv;
    const int b  = lin;
    const int woBase = wt * 16;

    const int tid  = threadIdx.x;
    const int lane = tid & 31;
    const int wave = tid >> 5;

    const float* __restrict__ xb = x + (size_t)b * CINv * HWv;

    // ---- Phase A: per-(k2,pixel) bilinear corner tables (mask folded in) ----
    if (tid < 144) {
        const int q  = tid;
        const int k2 = q >> 4;
        const int p  = q & 15;
        const int ki = k2 / 3, kj = k2 % 3;
        const int wo = woBase + p;

        const size_t pixOff = (size_t)ho * Wv + wo;
        const float o1 = offsets[((size_t)b * 18 + k2)     * HWv + pixOff];
        const float o2 = offsets[((size_t)b * 18 + 9 + k2) * HWv + pixOff];
        const float mk = mask   [((size_t)b * 9  + k2)     * HWv + pixOff];

        const float py = (float)(ho - 1 + ki) + o1;
        const float px = (float)(wo - 1 + kj) + o2;
        const float y0f = floorf(py), x0f = floorf(px);
        const float ly = py - y0f, lx = px - x0f;
        const int y0 = (int)y0f, x0 = (int)x0f;
        const int y1 = y0 + 1,   x1 = x0 + 1;

        auto corner = [&](int yi, int xi, float w, int slot) {
            const bool valid = (yi >= 0) && (yi < Hv) && (xi >= 0) && (xi < Wv);
            const int yc = min(max(yi, 0), Hv - 1);
            const int xc = min(max(xi, 0), Wv - 1);
            sIdx[slot * 144 + q] = yc * Wv + xc;
            sWgt[slot * 144 + q] = valid ? (w * mk) : 0.0f;
        };
        corner(y0, x0, (1.0f - ly) * (1.0f - lx), 0);
        corner(y0, x1, (1.0f - ly) * lx,          1);
        corner(y1, x0, ly * (1.0f - lx),          2);
        corner(y1, x1, ly * lx,                   3);
    }
    __syncthreads();

    const int m16   = lane & 15;
    const int khalf = (lane >> 4) << 1;           // 0 or 2
    const int coBase = wave * 16;
    const float* __restrict__ wrow = weight + (size_t)(coBase + m16) * REDv + khalf;

    v8f acc = {};

    for (int half = 0; half < 2; ++half) {
        if (half) __syncthreads();   // WAR: previous WMMA reads of sB finished

        // ---- Phase B: gather 64 channels x 9 taps x 16 pixels into sB ----
        for (int e = tid; e < 64 * 144; e += 256) {
            const int cl = e / 144;              // local channel 0..63
            const int q  = e % 144;              // k2*16 + p
            const int c  = half * 64 + cl;
            const float* __restrict__ xc = xb + (size_t)c * HWv;
            float s = sWgt[0 * 144 + q] * xc[sIdx[0 * 144 + q]];
            s = fmaf(sWgt[1 * 144 + q], xc[sIdx[1 * 144 + q]], s);
            s = fmaf(sWgt[2 * 144 + q], xc[sIdx[2 * 144 + q]], s);
            s = fmaf(sWgt[3 * 144 + q], xc[sIdx[3 * 144 + q]], s);
            const int rloc = cl * 9 + (q >> 4);  // local reduction row
            sB[rloc * 16 + (q & 15)] = s;
        }
        __syncthreads();

        // ---- Phase C: 144 fp32 WMMAs over this half of the reduction ----
        const float* __restrict__ wrh = wrow + half * HALF_RED;
        for (int kk = 0; kk < HALF_RED; kk += 4) {
            v2f a, bf;
            a.x  = wrh[kk];
            a.y  = wrh[kk + 1];
            bf.x = sB[(kk + khalf)     * 16 + m16];
            bf.y = sB[(kk + khalf + 1) * 16 + m16];
            acc = __builtin_amdgcn_wmma_f32_16x16x4_f32(
                false, a, false, bf, (short)0, acc, false, false);
        }
    }

    // ---- Epilogue: bias + store ----
    const int hi = lane >> 4;                    // row-half select
    float* __restrict__ outp =
        out + (size_t)b * COUTv * HWv + (size_t)ho * Wv + (woBase + m16);
#pragma unroll
    for (int v = 0; v < 8; ++v) {
        const int co = coBase + hi * 8 + v;
        outp[(size_t)co * HWv] = acc[v] + bias[co];
    }
}

// ---------------------------------------------------------------------------
extern "C" void kernel_launch(void* const* d_in, const int* in_sizes, int n_in,
                              void* d_out, int out_size, void* d_ws, size_t ws_size,
                              hipStream_t stream) {
    (void)in_sizes; (void)n_in; (void)out_size; (void)ws_size;

    const float* x         = (const float*)d_in[0];  // [4,128,96,96]
    const float* weight    = (const float*)d_in[1];  // [128,128,3,3]
    const float* bias      = (const float*)d_in[2];  // [128]
    const float* w_offmask = (const float*)d_in[3];  // [27,128,3,3]

    float* out        = (float*)d_out;                          // [4,128,96,96]
    float* offset_out = out + (size_t)Bv * COUTv * HWv;         // [4,18,96,96]
    float* mask_ws    = (float*)d_ws;                           // [4,9,96,96] (1.33 MB)

    const int tiles = Bv * Hv * (Wv / 16);   // 2304

    // Kernel 1: offset/mask conv via im2col + WMMA
    offmask_wmma_kernel<<<dim3(tiles), dim3(256), 0, stream>>>(
        x, w_offmask, offset_out, mask_ws);

    // Kernel 2: deformable sampling + fp32 WMMA GEMM
    deform_wmma_kernel<<<dim3(tiles), dim3(256), 0, stream>>>(
        x, weight, bias, offset_out, mask_ws, out);
}